// reparameter_encoder_42769284333873
// MI455X (gfx1250) — compile-verified
//
#include <hip/hip_runtime.h>
#include <stdint.h>

#define NN      4096
#define BB      32
#define TT      600
#define THREADS 512
#define REPS    4
#define KV      4                // float2 groups per thread (8 elems)
#define WAVES   (THREADS / 32)   // 16

typedef unsigned int uv4 __attribute__((ext_vector_type(4)));
typedef int          iv8 __attribute__((ext_vector_type(8)));
typedef int          iv4 __attribute__((ext_vector_type(4)));
typedef float        f2  __attribute__((ext_vector_type(2)));
typedef float        f4  __attribute__((ext_vector_type(4)));

// Hardware transcendental tanh (CDNA5 V_TANH_F32). Trailing v_nop satisfies the
// "1 independent op or V_NOP after TRANS before result use" hazard rule.
__device__ __forceinline__ float htanh(float x) {
    float y;
    asm("v_tanh_f32 %0, %1\n\tv_nop" : "=v"(y) : "v"(x));
    return y;
}

// Issue one TDM DMA: contiguous row of NN floats (16 KB) global -> LDS.
// D# per CDNA5 ISA 8.3/8.4; 6-arg builtin on this toolchain. Tracked by TENSORcnt.
__device__ __forceinline__ void tdm_load_row(uint32_t lds_byte_off, const float* gsrc) {
    uint64_t ga = (uint64_t)(uintptr_t)gsrc;
    uv4 g0;
    g0[0] = 1u;                                          // count=1, user mode, no gather
    g0[1] = lds_byte_off;                                // lds_addr
    g0[2] = (uint32_t)ga;                                // global_addr[31:0]
    g0[3] = (uint32_t)((ga >> 32) & 0x01FFFFFFu)         // global_addr[56:32]
          | (2u << 30);                                  // type=2 ("image")
    iv8 g1;
    g1[0] = 0x00020000;                                  // data_size=2 -> 4 bytes
    g1[1] = (int)((NN & 0xFFFF) << 16);                  // tensor_dim0[15:0]
    g1[2] = (int)((NN >> 16) | (1 << 16));               // tensor_dim0[31:16] | tensor_dim1=1
    g1[3] = (int)((NN & 0xFFFF) << 16);                  // tile_dim0=NN
    g1[4] = 1;                                           // tile_dim1=1, tile_dim2=0
    g1[5] = NN;                                          // tensor_dim0_stride[31:0]
    g1[6] = (int)((NN & 0xFFFF) << 16);                  // tensor_dim1_stride[15:0]
    g1[7] = 0;
    iv4 z4 = {0, 0, 0, 0};
    iv8 z8 = {0, 0, 0, 0, 0, 0, 0, 0};
    __builtin_amdgcn_tensor_load_to_lds(g0, g1, z4, z4, z8, 0);
}

__global__ __launch_bounds__(THREADS)
void reparameter_encoder_42769284333873_kernel(
    const float* __restrict__ input,   // [B,T,1]
    const float* __restrict__ Imat,    // [N,1]
    const float* __restrict__ Umat,    // [N,2]
    const float* __restrict__ Vmat,    // [N,2]
    const float* __restrict__ nrec,    // [B,T,N]
    const float* __restrict__ ninp,    // [B,T,1]
    float* __restrict__ out)           // [B,T+1,N]
{
    const int tid  = threadIdx.x;
    const int b    = blockIdx.x & (BB - 1);
    const int rep  = blockIdx.x >> 5;          // 0..REPS-1
    const int wid  = tid >> 5;
    const int lane = tid & 31;

    __shared__ float nbuf[2][NN];              // double-buffered TDM target
    __shared__ float red[2 * WAVES];

    // Thread owns n = 2*tid + 1024*kk + j  (kk=0..3, j=0..1): pairwise contiguous.
    // Deinterleave U/V so the inner loop is pure packed-f32 FMAs.
    f2 Ua[KV], Ub[KV], Va[KV], Vb[KV], Iv[KV], c[KV];
    {
        const f4* U4 = (const f4*)Umat;
        const f4* V4 = (const f4*)Vmat;
        const f2* I2 = (const f2*)Imat;
#pragma unroll
        for (int kk = 0; kk < KV; ++kk) {
            int g = tid + 512 * kk;            // = (2n)/4 group index
            f4 u = U4[g], v = V4[g];
            Ua[kk] = (f2){u.x, u.z};           // U[:,0] for n, n+1
            Ub[kk] = (f2){u.y, u.w};           // U[:,1]
            Va[kk] = (f2){v.x, v.z};
            Vb[kk] = (f2){v.y, v.w};
            Iv[kk] = I2[g];
            c[kk]  = (f2){0.0f, 0.0f};
        }
    }

    const float al    = 0.1f, em = 0.9f;
    const float inv_n = 1.0f / (float)NN;
    const float s_rec = 0.022360679774997897f;   // alpha * sig_rec * sqrt(2*tau/dt)
    const float s_inp = 0.004472135954999579f;   // sig_inp * sqrt(2*dt/tau)

    f2* out2 = (f2*)out;

    // t = 0 output: h0 = 0; each block writes only its exclusive slice (kk == rep).
    {
        size_t base2 = (size_t)b * (TT + 1) * (NN / 2) + (size_t)(tid + 512 * rep);
        __builtin_nontemporal_store((f2){0.0f, 0.0f}, &out2[base2]);
    }

    const float* nrow = nrec + (size_t)b * TT * NN;

    // Prime the pipeline: wave 0 DMAs the t=1 noise row into buffer 0.
    if (wid == 0) tdm_load_row((uint32_t)(uintptr_t)&nbuf[0][0], nrow);

    float a0 = 0.0f, a1 = 0.0f, bsc = 0.0f;
    f2 pa = {0.0f, 0.0f}, pb = {0.0f, 0.0f};   // tanh(h_{t-1})·V partials (0 for t=1)
    int buf = 0;

    for (int t = 1; t <= TT; ++t) {
        // ---- kappa_t: block-wide reduction of previous step's partials ----
        float p0 = pa.x + pa.y;
        float p1 = pb.x + pb.y;
#pragma unroll
        for (int off = 16; off > 0; off >>= 1) {
            p0 += __shfl_xor(p0, off, 32);
            p1 += __shfl_xor(p1, off, 32);
        }
        if (lane == 0) { red[2 * wid] = p0; red[2 * wid + 1] = p1; }
        __syncthreads();                         // (A) red visible; prev buffer drained
        float S0 = 0.0f, S1 = 0.0f;
#pragma unroll
        for (int w = 0; w < WAVES; ++w) { S0 += red[2 * w]; S1 += red[2 * w + 1]; }
        a0 = em * a0 + al * (inv_n * S0);
        a1 = em * a1 + al * (inv_n * S1);

        float x  = input[(size_t)b * TT + (t - 1)];
        float nx = ninp [(size_t)b * TT + (t - 1)];
        bsc = em * bsc + al * (x + nx * s_inp);

        // ---- TDM pipeline: wait on row t, kick off row t+1 into the other buffer ----
        if (wid == 0) {
            __builtin_amdgcn_s_wait_tensorcnt(0);
            if (t < TT)
                tdm_load_row((uint32_t)(uintptr_t)&nbuf[buf ^ 1][0], nrow + (size_t)t * NN);
        }
        __syncthreads();                         // (B) nbuf[buf] holds noise for step t

        // ---- elementwise update over full N + partials for kappa_{t+1} ----
        pa = (f2){0.0f, 0.0f};
        pb = (f2){0.0f, 0.0f};
        size_t obase2 = ((size_t)b * (TT + 1) + t) * (NN / 2);
#pragma unroll
        for (int kk = 0; kk < KV; ++kk) {
            int n = 2 * tid + 1024 * kk;
            f2 xi = *(const f2*)&nbuf[buf][n];           // ds_load_b64, bank-conflict-free
            f2 cc = em * c[kk] + s_rec * xi;             // packed f32 FMAs
            c[kk] = cc;
            f2 h = a0 * Ua[kk] + a1 * Ub[kk] + bsc * Iv[kk] + cc;
            if (kk == rep)                               // exclusive output slice
                __builtin_nontemporal_store(h, &out2[obase2 + tid + 512 * rep]);
            f2 fr;
            fr.x = htanh(h.x);
            fr.y = htanh(h.y);
            pa += fr * Va[kk];
            pb += fr * Vb[kk];
        }
        buf ^= 1;
    }
}

extern "C" void kernel_launch(void* const* d_in, const int* in_sizes, int n_in,
                              void* d_out, int out_size, void* d_ws, size_t ws_size,
                              hipStream_t stream) {
    const float* input = (const float*)d_in[0];
    const float* Imat  = (const float*)d_in[1];
    const float* Umat  = (const float*)d_in[2];
    const float* Vmat  = (const float*)d_in[3];
    const float* nrec  = (const float*)d_in[4];
    const float* ninp  = (const float*)d_in[5];
    float* out = (float*)d_out;

    hipLaunchKernelGGL(reparameter_encoder_42769284333873_kernel,
                       dim3(BB * REPS), dim3(THREADS), 0, stream,
                       input, Imat, Umat, Vmat, nrec, ninp, out);
}